// AdditiveAttention_22265110463295
// MI455X (gfx1250) — compile-verified
//
#include <hip/hip_runtime.h>
#include <hip/hip_bf16.h>
#include <math.h>

// ---------------------------------------------------------------------------
// Additive attention, fused, for MI455X (gfx1250, wave32).
//   B=4, n=512, m=1024, d=256, h=128, dv=256
// Kernel 1: fp32 projection GEMM via V_WMMA_F32_16X16X4_F32 (tq, tk)
// Kernel 2: fused tanh-score + online softmax + (P @ V) via f32 WMMA
// ---------------------------------------------------------------------------

typedef __attribute__((ext_vector_type(2))) float v2f;
typedef __attribute__((ext_vector_type(8))) float v8f;

#define ATT_B  4
#define ATT_N  512
#define ATT_M  1024
#define ATT_D  256
#define ATT_H  128
#define ATT_DV 256
#define CM     32          // m-chunk per fused-kernel iteration
#define TK_PAD 132         // LDS row stride for tk tile (2-way conflict max)
#define P_PAD  33          // LDS row stride for probability tile

__device__ __forceinline__ float fast_tanh(float x) {
#if __has_builtin(__builtin_amdgcn_tanhf)
    return __builtin_amdgcn_tanhf(x);
#else
    return tanhf(x);
#endif
}

// ---------------------------------------------------------------------------
// Projection: Y[R x H] = X[R x D] @ W[D x H] + bias[H]
// One wave per 16x16 output tile; K loop in steps of 4 using f32 WMMA.
// ---------------------------------------------------------------------------
__global__ void proj_wmma_f32(const float* __restrict__ X,
                              const float* __restrict__ W,
                              const float* __restrict__ bias,
                              float* __restrict__ Y,
                              int R) {
    const int D = ATT_D;
    const int H = ATT_H;
    const int wavesPerBlock = blockDim.x >> 5;
    const int wave = blockIdx.x * wavesPerBlock + (threadIdx.x >> 5);
    const int ntilesN = H / 16;           // 8
    const int mt = wave / ntilesN;
    const int nt = wave % ntilesN;
    if (mt * 16 >= R) return;             // whole-wave uniform (R % 16 == 0)

    const int lane = threadIdx.x & 31;
    const int half = lane >> 4;
    const int l16  = lane & 15;
    const int arow = mt * 16 + l16;
    const int bcol = nt * 16 + l16;

    v8f c = {};
#if __has_builtin(__builtin_amdgcn_wmma_f32_16x16x4_f32)
    #pragma unroll 4
    for (int k = 0; k < D; k += 4) {
        const int kb = k + 2 * half;
        v2f a, b;
        a.x = X[arow * D + kb];
        a.y = X[arow * D + kb + 1];
        b.x = W[kb * H + bcol];
        b.y = W[(kb + 1) * H + bcol];
        c = __builtin_amdgcn_wmma_f32_16x16x4_f32(
                false, a, false, b, (short)0, c, false, false);
    }
#else
    for (int r = 0; r < 8; ++r) {
        const int row = mt * 16 + r + 8 * half;
        float acc = 0.f;
        for (int k = 0; k < D; ++k) acc += X[row * D + k] * W[k * H + bcol];
        c[r] = acc;
    }
#endif

    const float bb = bias[bcol];
    #pragma unroll
    for (int r = 0; r < 8; ++r) {
        const int row = mt * 16 + r + 8 * half;
        Y[row * H + bcol] = c[r] + bb;
    }
}

// ---------------------------------------------------------------------------
// Fused kernel: one block = 16 query rows, 512 threads = 16 wave32.
// Score phase : wave w owns query row w, lanes cover a 32-wide m-chunk.
// PV phase    : wave w owns dv columns [16w, 16w+16), 8x f32 WMMA per chunk.
// ---------------------------------------------------------------------------
__global__ void __launch_bounds__(512)
attn_fused(const float* __restrict__ tq,     // (B*N, H)
           const float* __restrict__ tk,     // (B*M, H)
           const float* __restrict__ values, // (B, M, DV)
           const int*   __restrict__ mask,   // (B, N, M)
           const float* __restrict__ wv,     // (H)
           const float* __restrict__ bv,     // scalar
           float* __restrict__ out)          // (B, N, DV)
{
    const int qbase = blockIdx.x * 16;       // global query-row base (0..2047)
    const int b     = qbase / ATT_N;         // batch (tiles never straddle)
    const int tid   = threadIdx.x;           // 0..511
    const int lane  = tid & 31;
    const int wid   = tid >> 5;              // 0..15
    const int half  = lane >> 4;
    const int l16   = lane & 15;

    __shared__ float s_tq[16 * ATT_H];       // 8 KB (broadcast reads)
    __shared__ float s_wv[ATT_H];
    __shared__ float s_tk[CM * TK_PAD];      // ~16.5 KB, padded
    __shared__ float s_p[16 * P_PAD];        // padded probability tile
    __shared__ float s_scale[16];
    __shared__ float s_sum[16];

    // stage tq tile (2048 floats = 512 float4) + wv
    ((float4*)s_tq)[tid] = ((const float4*)(tq + (size_t)qbase * ATT_H))[tid];
    if (tid < ATT_H) s_wv[tid] = wv[tid];
    __syncthreads();

    const float  bias_v  = bv[0];
    const float* tkb     = tk + (size_t)b * ATT_M * ATT_H;
    const float* vb      = values + (size_t)b * ATT_M * ATT_DV;
    const int*   maskrow = mask + ((size_t)qbase + wid) * ATT_M;

    float run_max = -INFINITY;
    float run_sum = 0.0f;
    v8f c = {};

    const int stg_i = tid >> 4;              // 0..31 : tk row within chunk
    const int stg_h = (tid & 15) * 8;        // 0..120: 8 floats per thread

    for (int m0 = 0; m0 < ATT_M; m0 += CM) {
        // ---- stage tk chunk (32 x 128 f32) into padded LDS ----------------
        {
            const float* src = tkb + (size_t)(m0 + stg_i) * ATT_H + stg_h;
            const float4 a0 = ((const float4*)src)[0];
            const float4 a1 = ((const float4*)src)[1];
            *(float4*)(s_tk + stg_i * TK_PAD + stg_h)     = a0;
            *(float4*)(s_tk + stg_i * TK_PAD + stg_h + 4) = a1;
            if (m0 + CM < ATT_M)   // touch next chunk (global_prefetch_b8)
                __builtin_prefetch(src + (size_t)CM * ATT_H, 0, 1);
        }
        __syncthreads();

        // ---- score: row = wid, m = m0 + lane ------------------------------
        float s = 0.0f;
        {
            const float* tkr = s_tk + lane * TK_PAD;
            const float* tqr = s_tq + wid * ATT_H;
            #pragma unroll 4
            for (int h = 0; h < ATT_H; h += 4) {
                const float4 kv = *(const float4*)(tkr + h);
                s += s_wv[h + 0] * fast_tanh(tqr[h + 0] + kv.x);
                s += s_wv[h + 1] * fast_tanh(tqr[h + 1] + kv.y);
                s += s_wv[h + 2] * fast_tanh(tqr[h + 2] + kv.z);
                s += s_wv[h + 3] * fast_tanh(tqr[h + 3] + kv.w);
            }
        }
        s += bias_v;
        if (maskrow[m0 + lane] == 0) s = -1.0e10f;

        // ---- wave32 butterfly reductions: all lanes get max & sum ---------
        float cmax = s;
        #pragma unroll
        for (int off = 16; off > 0; off >>= 1)
            cmax = fmaxf(cmax, __shfl_xor(cmax, off, 32));
        const float new_max = fmaxf(run_max, cmax);
        const float p = __expf(s - new_max);
        float csum = p;
        #pragma unroll
        for (int off = 16; off > 0; off >>= 1)
            csum += __shfl_xor(csum, off, 32);
        const float scale = __expf(run_max - new_max);   // 0 on first chunk
        run_sum = run_sum * scale + csum;
        run_max = new_max;

        s_p[wid * P_PAD + lane] = p;
        if (lane == 0) s_scale[wid] = scale;
        __syncthreads();

        // ---- PV: wave wid owns cols [16*wid, 16*wid+16) -------------------
        const int col = wid * 16 + l16;
        #pragma unroll
        for (int r = 0; r < 8; ++r) c[r] *= s_scale[r + 8 * half];

#if __has_builtin(__builtin_amdgcn_wmma_f32_16x16x4_f32)
        #pragma unroll
        for (int kk = 0; kk < CM; kk += 4) {
            v2f a, bmat;
            // A 16x4: lane row = l16, K = kk + 2*half (+1)
            a.x = s_p[l16 * P_PAD + kk + 2 * half];
            a.y = s_p[l16 * P_PAD + kk + 2 * half + 1];
            // B 4x16: V rows (m0+kk+2*half, +1), col = 16*wid + l16
            const size_t vrow = (size_t)(m0 + kk + 2 * half);
            bmat.x = vb[vrow * ATT_DV + col];
            bmat.y = vb[(vrow + 1) * ATT_DV + col];
            c = __builtin_amdgcn_wmma_f32_16x16x4_f32(
                    false, a, false, bmat, (short)0, c, false, false);
        }
#else
        #pragma unroll
        for (int r = 0; r < 8; ++r) {
            const int row = r + 8 * half;
            float acc = c[r];
            for (int j = 0; j < CM; ++j)
                acc += s_p[row * P_PAD + j] * vb[(size_t)(m0 + j) * ATT_DV + col];
            c[r] = acc;
        }
#endif
        __syncthreads();
    }

    if (lane == 0) s_sum[wid] = run_sum;
    __syncthreads();

    // C/D 16x16 f32: VGPR r -> tile row (r + 8*half), col = l16
    #pragma unroll
    for (int r = 0; r < 8; ++r) {
        const int row = r + 8 * half;
        out[((size_t)qbase + row) * ATT_DV + wid * 16 + l16] = c[r] / s_sum[row];
    }
}

// ---------------------------------------------------------------------------
extern "C" void kernel_launch(void* const* d_in, const int* in_sizes, int n_in,
                              void* d_out, int out_size, void* d_ws, size_t ws_size,
                              hipStream_t stream) {
    const float* queries = (const float*)d_in[0];   // (B, N, D)
    const float* keys    = (const float*)d_in[1];   // (B, M, D)
    const float* values  = (const float*)d_in[2];   // (B, M, DV)
    const int*   mask    = (const int*)  d_in[3];   // (B, N, M)
    const float* Wq      = (const float*)d_in[4];   // (D, H)
    const float* bq      = (const float*)d_in[5];   // (H)
    const float* Wk      = (const float*)d_in[6];   // (D, H)
    const float* bk      = (const float*)d_in[7];   // (H)
    const float* wv      = (const float*)d_in[8];   // (H)
    const float* bv      = (const float*)d_in[9];   // scalar
    float* out = (float*)d_out;

    float* tq = (float*)d_ws;                         // (B*N, H) = 2048*128
    float* tk = tq + (size_t)ATT_B * ATT_N * ATT_H;   // (B*M, H) = 4096*128

    const int RQ = ATT_B * ATT_N;   // 2048
    const int RK = ATT_B * ATT_M;   // 4096
    const int ntilesN = ATT_H / 16; // 8
    const int wavesPerBlock = 4;    // 128 threads

    proj_wmma_f32<<<(RQ / 16) * ntilesN / wavesPerBlock, 32 * wavesPerBlock,
                    0, stream>>>(queries, Wq, bq, tq, RQ);
    proj_wmma_f32<<<(RK / 16) * ntilesN / wavesPerBlock, 32 * wavesPerBlock,
                    0, stream>>>(keys, Wk, bk, tk, RK);

    // fused score/softmax/PV: one block per 16 query rows
    attn_fused<<<ATT_B * ATT_N / 16, 512, 0, stream>>>(
        tq, tk, values, mask, wv, bv, out);
}